// DecompositionBlock_14096082666335
// MI455X (gfx1250) — compile-verified
//
#include <hip/hip_runtime.h>

// ---------------------------------------------------------------------------
// DecompositionBlock for MI455X (gfx1250, wave32, WMMA f32 16x16x4)
//
// reference: x:(4,256,64,256) f32, p:(256,32) f32
//   Haar wavedec (2 levels) along T -> 3 scales (L=128,64,64)
//   per (scale,b,l): A=(64x256); z=A@p; q=QR(z).Q; y=q^T A;
//   top-2 SVD of y -> terms s_j u_j vh_j ; residual = A - terms
//   output: (9,4,64,256,256) f32, left-padded along last dim
//
// term_j == (q u_j)(u_j^T y) with u_j = top eigenvectors of y y^T
// (singular value cancels; invariant to QR/SVD sign conventions)
// ---------------------------------------------------------------------------

#define B_ 4
#define T_ 256
#define C_ 64
#define N_ 256
#define K_ 32
#define INV_SQRT2 0.70710678118654752f

typedef float v2f __attribute__((ext_vector_type(2)));
typedef float v8f __attribute__((ext_vector_type(8)));

// workspace layout (floats): uf[m][2][64], w[m][2][256], m = 0..1023
#define UF_OFF 0
#define W_OFF  131072          // 1024*2*64
// total ws use: 655360 floats = 2.62 MB

// phase-2 dynamic LDS layout (floats); odd-ish strides -> conflict-free cols
#define SA_STR 265
#define SY_STR 265
#define OFF_A  0
#define OFF_Y  (64 * SA_STR)               // 16960
#define OFF_Z  (OFF_Y + 32 * SY_STR)       // 25440
#define OFF_Q  (OFF_Z + 64 * 33)           // 27552
#define OFF_G  (OFF_Q + 64 * 33)           // 29664  (G, then Rinv, then G2)
#define OFF_R  (OFF_G + 32 * 33)           // 30720
#define OFF_U0 (OFF_R + 32 * 33)           // 31776
#define OFF_U1 (OFF_U0 + 32)
#define SMEM_FLOATS (OFF_U1 + 64)          // 31872 floats = 127488 B

// ---- WMMA helpers: V_WMMA_F32_16X16X4_F32, layouts per ISA 7.12.2 ----------
__device__ __forceinline__ v8f wmma4(v2f a, v2f b, v8f c) {
  // 8 args: (neg_a, A, neg_b, B, c_mod, C, reuse_a, reuse_b)
  return __builtin_amdgcn_wmma_f32_16x16x4_f32(false, a, false, b, (short)0, c,
                                               false, false);
}
// A (16x4) row-major source S[m][k]: lane m0+lm, VGPR0=K(2h), VGPR1=K(2h+1)
__device__ __forceinline__ v2f fragA_rm(const float* S, int str, int m0, int k0,
                                        int lm, int h) {
  v2f a;
  a.x = S[(m0 + lm) * str + k0 + 2 * h];
  a.y = S[(m0 + lm) * str + k0 + 2 * h + 1];
  return a;
}
// A (16x4) where A[m][k] = S[k][m] (transposed read)
__device__ __forceinline__ v2f fragA_tr(const float* S, int str, int m0, int k0,
                                        int lm, int h) {
  v2f a;
  a.x = S[(k0 + 2 * h) * str + m0 + lm];
  a.y = S[(k0 + 2 * h + 1) * str + m0 + lm];
  return a;
}
// B (4x16) row-major source S[k][n]
__device__ __forceinline__ v2f fragB_rm(const float* S, int str, int k0, int n0,
                                        int lm, int h) {
  v2f b;
  b.x = S[(k0 + 2 * h) * str + n0 + lm];
  b.y = S[(k0 + 2 * h + 1) * str + n0 + lm];
  return b;
}
// B (4x16) where B[k][n] = S[n][k] (transposed read)
__device__ __forceinline__ v2f fragB_tr(const float* S, int str, int k0, int n0,
                                        int lm, int h) {
  v2f b;
  b.x = S[(n0 + lm) * str + k0 + 2 * h];
  b.y = S[(n0 + lm) * str + k0 + 2 * h + 1];
  return b;
}
// C/D (16x16): VGPR v -> row m0+v+8h, col n0+lm
__device__ __forceinline__ void storeC(float* D, int str, int m0, int n0,
                                       int lm, int h, v8f c) {
#pragma unroll
  for (int vg = 0; vg < 8; ++vg) D[(m0 + vg + 8 * h) * str + n0 + lm] = c[vg];
}

__device__ __forceinline__ void scale_taps(int s, int& nt, int& tsc, float& h0,
                                           float& h1, float& h2, float& h3) {
  if (s == 0)      { nt = 2; tsc = 2; h0 =  INV_SQRT2; h1 = -INV_SQRT2; h2 = 0.f;   h3 = 0.f;   }
  else if (s == 1) { nt = 4; tsc = 4; h0 =  0.5f;      h1 =  0.5f;      h2 = -0.5f; h3 = -0.5f; }
  else             { nt = 4; tsc = 4; h0 =  0.5f;      h1 =  0.5f;      h2 =  0.5f; h3 =  0.5f; }
}

// ===========================================================================
// Phase 2: per-matrix sketch + CholeskyQR + Gram eigen; writes uf/w factors
// ===========================================================================
__global__ void __launch_bounds__(256) decomp_phase2(
    const float* __restrict__ x, const float* __restrict__ p,
    float* __restrict__ ws) {
  extern __shared__ float sm[];
  float* sA  = sm + OFF_A;   // 64 x 256  (stride 265)
  float* sY  = sm + OFF_Y;   // 32 x 256  (stride 265)
  float* sZ  = sm + OFF_Z;   // 64 x 32   (stride 33)
  float* sQ  = sm + OFF_Q;   // 64 x 32   (stride 33)
  float* sG  = sm + OFF_G;   // 32 x 32   (stride 33): G -> Rinv -> G2
  float* sR  = sm + OFF_R;   // 32 x 32   (stride 33)
  float* sU0 = sm + OFF_U0;
  float* sU1 = sm + OFF_U1;

  const int tid  = threadIdx.x;
  const int wave = tid >> 5;
  const int lane = tid & 31;
  const int lm   = lane & 15;
  const int h    = lane >> 4;

  // decode matrix index m -> (scale s, batch b, position l)
  const int m = blockIdx.x;
  int s, b, l;
  if (m < 512)      { s = 0; b = m >> 7;         l = m & 127; }
  else if (m < 768) { s = 1; b = (m - 512) >> 6; l = (m - 512) & 63; }
  else              { s = 2; b = (m - 768) >> 6; l = (m - 768) & 63; }
  int nt, tsc; float h0, h1, h2, h3;
  scale_taps(s, nt, tsc, h0, h1, h2, h3);
  const int tb = l * tsc;

  // ---- A(64x256) into LDS, Haar computed on the fly from x ----------------
  // x index: ((b*T + t)*C + c)*N + n ; tid sweeps n (coalesced rows)
  {
    const size_t xb = ((size_t)b * T_ + tb) * (size_t)(C_ * N_);
    for (int c = 0; c < C_; ++c) {
      const size_t rb = xb + (size_t)c * N_;
      if (c + 8 < C_) __builtin_prefetch(&x[rb + 8 * N_ + tid], 0, 1);
      float acc = h0 * x[rb + tid] + h1 * x[rb + (size_t)(C_ * N_) + tid];
      if (nt == 4)
        acc += h2 * x[rb + (size_t)(2 * C_ * N_) + tid] +
               h3 * x[rb + (size_t)(3 * C_ * N_) + tid];
      sA[c * SA_STR + tid] = acc;
    }
  }
  __syncthreads();

  // ---- GEMM1: z = A(64x256) @ p(256x32) ; 8 tiles = 8 waves ---------------
  {
    const int m0 = (wave >> 1) * 16, n0 = (wave & 1) * 16;
    v8f acc = {};
    for (int k0 = 0; k0 < 256; k0 += 4) {
      v2f a = fragA_rm(sA, SA_STR, m0, k0, lm, h);
      v2f bb;  // p row-major [k][n], stride 32 (global, L2-hot)
      bb.x = p[(k0 + 2 * h) * K_ + n0 + lm];
      bb.y = p[(k0 + 2 * h + 1) * K_ + n0 + lm];
      acc = wmma4(a, bb, acc);
    }
    storeC(sZ, 33, m0, n0, lm, h, acc);
  }
  __syncthreads();

  // ---- G = z^T z (32x32) ; 4 tiles on waves 0..3 --------------------------
  if (wave < 4) {
    const int m0 = (wave >> 1) * 16, n0 = (wave & 1) * 16;
    v8f acc = {};
    for (int k0 = 0; k0 < 64; k0 += 4) {
      v2f a  = fragA_tr(sZ, 33, m0, k0, lm, h);
      v2f bb = fragB_rm(sZ, 33, k0, n0, lm, h);
      acc = wmma4(a, bb, acc);
    }
    storeC(sG, 33, m0, n0, lm, h, acc);
  }
  __syncthreads();

  // ---- Cholesky G = R^T R (upper R), then Rinv = R^{-1} into sG -----------
  if (tid < 32) {
    const int j = lane;
    for (int i = 0; i < 32; ++i) {
      float acc = sG[i * 33 + j];
      for (int k = 0; k < i; ++k) acc -= sR[k * 33 + i] * sR[k * 33 + j];
      float d = __shfl(acc, i, 32);             // diagonal candidate
      d = sqrtf(fmaxf(d, 1e-20f));
      float rij = (j > i) ? (acc / d) : ((j == i) ? d : 0.0f);
      sR[i * 33 + j] = rij;
    }
    // back-substitution: column j of R^{-1} (upper) -> sG
    sG[j * 33 + j] = 1.0f / sR[j * 33 + j];
    for (int i = j - 1; i >= 0; --i) {
      float acc2 = 0.0f;
      for (int k = i + 1; k <= j; ++k) acc2 += sR[i * 33 + k] * sG[k * 33 + j];
      sG[i * 33 + j] = -acc2 / sR[i * 33 + i];
    }
    for (int i = j + 1; i < 32; ++i) sG[i * 33 + j] = 0.0f;
  }
  __syncthreads();

  // ---- q = z(64x32) @ Rinv(32x32) ; 8 tiles = 8 waves ---------------------
  {
    const int m0 = (wave >> 1) * 16, n0 = (wave & 1) * 16;
    v8f acc = {};
    for (int k0 = 0; k0 < 32; k0 += 4) {
      v2f a  = fragA_rm(sZ, 33, m0, k0, lm, h);
      v2f bb = fragB_rm(sG, 33, k0, n0, lm, h);
      acc = wmma4(a, bb, acc);
    }
    storeC(sQ, 33, m0, n0, lm, h, acc);
  }
  __syncthreads();

  // ---- y = q^T(32x64) @ A(64x256) ; 32 tiles, 4 per wave ------------------
  for (int t4 = 0; t4 < 4; ++t4) {
    const int tile = wave + 8 * t4;
    const int m0 = (tile >> 4) * 16, n0 = (tile & 15) * 16;
    v8f acc = {};
    for (int k0 = 0; k0 < 64; k0 += 4) {
      v2f a  = fragA_tr(sQ, 33, m0, k0, lm, h);          // q^T[m][k] = q[k][m]
      v2f bb = fragB_rm(sA, SA_STR, k0, n0, lm, h);
      acc = wmma4(a, bb, acc);
    }
    storeC(sY, SY_STR, m0, n0, lm, h, acc);
  }
  __syncthreads();

  // ---- G2 = y y^T (32x32) ; 4 tiles on waves 0..3 -------------------------
  if (wave < 4) {
    const int m0 = (wave >> 1) * 16, n0 = (wave & 1) * 16;
    v8f acc = {};
    for (int k0 = 0; k0 < 256; k0 += 4) {
      v2f a  = fragA_rm(sY, SY_STR, m0, k0, lm, h);
      v2f bb = fragB_tr(sY, SY_STR, k0, n0, lm, h);      // y^T read
      acc = wmma4(a, bb, acc);
    }
    storeC(sG, 33, m0, n0, lm, h, acc);
  }
  __syncthreads();

  // ---- top-2 eigenvectors of G2 by power iteration + deflation (wave 0) ---
  if (tid < 32) {
    const int i = lane;
    float v = 1.0f + 0.017f * (float)i;
    for (int it = 0; it < 20; ++it) {
      float mv = 0.0f;
      for (int jj = 0; jj < 32; ++jj) mv += sG[i * 33 + jj] * __shfl(v, jj, 32);
      float ss = mv * mv;
      for (int o = 16; o > 0; o >>= 1) ss += __shfl_xor(ss, o, 32);
      v = mv * rsqrtf(ss + 1e-30f);
    }
    float mv = 0.0f;
    for (int jj = 0; jj < 32; ++jj) mv += sG[i * 33 + jj] * __shfl(v, jj, 32);
    float lam = v * mv;
    for (int o = 16; o > 0; o >>= 1) lam += __shfl_xor(lam, o, 32);
    sU0[i] = v;
    for (int jj = 0; jj < 32; ++jj)            // deflate
      sG[i * 33 + jj] -= lam * v * __shfl(v, jj, 32);
    float v2 = 1.0f - 0.013f * (float)i;
    {  // orthogonalize init against u0
      float d = v2 * v;
      for (int o = 16; o > 0; o >>= 1) d += __shfl_xor(d, o, 32);
      v2 -= d * v;
    }
    for (int it = 0; it < 20; ++it) {
      float mv2 = 0.0f;
      for (int jj = 0; jj < 32; ++jj) mv2 += sG[i * 33 + jj] * __shfl(v2, jj, 32);
      float ss = mv2 * mv2;
      for (int o = 16; o > 0; o >>= 1) ss += __shfl_xor(ss, o, 32);
      v2 = mv2 * rsqrtf(ss + 1e-30f);
    }
    sU1[i] = v2;
  }
  __syncthreads();

  // ---- factors: uf_j = q u_j (64) ; w_j = u_j^T y (256) -------------------
  if (tid < 128) {
    const int j = tid >> 6;
    const int c = tid & 63;
    const float* u = j ? sU1 : sU0;
    float acc = 0.0f;
    for (int i = 0; i < 32; ++i) acc += sQ[c * 33 + i] * u[i];
    ws[UF_OFF + ((size_t)m * 2 + j) * 64 + c] = acc;
  }
  {
    const int n = tid;
    float a0 = 0.0f, a1 = 0.0f;
    for (int i = 0; i < 32; ++i) {
      const float yv = sY[i * SY_STR + n];
      a0 += sU0[i] * yv;
      a1 += sU1[i] * yv;
    }
    ws[W_OFF + ((size_t)m * 2 + 0) * 256 + n] = a0;
    ws[W_OFF + ((size_t)m * 2 + 1) * 256 + n] = a1;
  }
}

// ===========================================================================
// Phase 3: expand rank-1 terms + residual, transpose in LDS, coalesced writes
// ===========================================================================
#define T3_STR 133
__global__ void __launch_bounds__(256) decomp_phase3(
    const float* __restrict__ x, const float* __restrict__ ws,
    float* __restrict__ out) {
  __shared__ float tile[64 * T3_STR];
  const int tid = threadIdx.x;
  const int bid = blockIdx.x;
  const int s  = bid >> 10;           // 0..2
  const int r  = bid & 1023;
  const int b  = r >> 8;              // 0..3
  const int c  = (r >> 2) & 63;       // 0..63
  const int n0 = (r & 3) * 64;

  const int L   = (s == 0) ? 128 : 64;
  const int pad = 256 - L;
  const int mbase = (s == 0) ? (b * 128) : ((s == 1) ? (512 + b * 64) : (768 + b * 64));
  int nt, tsc; float h0, h1, h2, h3;
  scale_taps(s, nt, tsc, h0, h1, h2, h3);

  const int lsub = tid >> 6;          // 0..3
  const int nn   = tid & 63;
  const int n    = n0 + nn;

  for (int idx = 0; idx < 3; ++idx) {
    for (int lb = 0; lb < L; lb += 4) {
      const int l  = lb + lsub;
      const int mm = mbase + l;
      float val;
      if (idx < 2) {
        val = ws[UF_OFF + ((size_t)mm * 2 + idx) * 64 + c] *
              ws[W_OFF + ((size_t)mm * 2 + idx) * 256 + n];
      } else {
        const int tb = l * tsc;
        const size_t rb = (((size_t)b * T_ + tb) * C_ + c) * (size_t)N_;
        float coeff = h0 * x[rb + n] + h1 * x[rb + (size_t)(C_ * N_) + n];
        if (nt == 4)
          coeff += h2 * x[rb + (size_t)(2 * C_ * N_) + n] +
                   h3 * x[rb + (size_t)(3 * C_ * N_) + n];
        coeff -= ws[UF_OFF + ((size_t)mm * 2 + 0) * 64 + c] *
                 ws[W_OFF + ((size_t)mm * 2 + 0) * 256 + n];
        coeff -= ws[UF_OFF + ((size_t)mm * 2 + 1) * 64 + c] *
                 ws[W_OFF + ((size_t)mm * 2 + 1) * 256 + n];
        val = coeff;
      }
      tile[nn * T3_STR + l] = val;
    }
    __syncthreads();
    // coalesced write: each padded 256-float output row in one sweep
    const int outIdx = s * 3 + idx;
    const size_t ob =
        ((((size_t)outIdx * B_ + b) * C_ + c) * N_ + n0) * (size_t)256;
    for (int rr = 0; rr < 64; ++rr) {
      const float vv = (tid >= pad) ? tile[rr * T3_STR + (tid - pad)] : 0.0f;
      out[ob + (size_t)rr * 256 + tid] = vv;
    }
    __syncthreads();
  }
}

// ===========================================================================
extern "C" void kernel_launch(void* const* d_in, const int* in_sizes, int n_in,
                              void* d_out, int out_size, void* d_ws,
                              size_t ws_size, hipStream_t stream) {
  const float* x = (const float*)d_in[0];   // (4,256,64,256) f32
  const float* p = (const float*)d_in[1];   // (256,32) f32
  float* ws  = (float*)d_ws;                // 2.62 MB used
  float* out = (float*)d_out;               // (9,4,64,256,256) f32

  const size_t smem = SMEM_FLOATS * sizeof(float);  // ~124.5 KB (CDNA5 WGP LDS)
  decomp_phase2<<<1024, 256, smem, stream>>>(x, p, ws);
  decomp_phase3<<<3072, 256, 0, stream>>>(x, ws, out);
}